// IttiKochSaliency_81114752352707
// MI455X (gfx1250) — compile-verified
//
#include <hip/hip_runtime.h>

#define NIMG 24              // B*C = 8*3 images
#define BIG 3.402823466e38f

typedef float v2f __attribute__((ext_vector_type(2)));
typedef float v8f __attribute__((ext_vector_type(8)));

// combined pyrdown 1-D weights: conv([1,4,6,4,1]/16,[1,1]/2) = [1,5,10,10,5,1]/32
__device__ __forceinline__ float wband(int idx){
  float v = 0.0f;
  v = (idx == 0 || idx == 5) ? 1.0f  : v;
  v = (idx == 1 || idx == 4) ? 5.0f  : v;
  v = (idx == 2 || idx == 3) ? 10.0f : v;
  return v * (1.0f/32.0f);
}

struct FinalArgs { long doff[7]; };

// ---------- per-image min/max over n elements ----------
__global__ void minmax_kernel(const float* __restrict__ x, float* __restrict__ stats, int n){
  const int tid = threadIdx.x;
  const float* p = x + (size_t)blockIdx.x * n;
  float mn = BIG, mx = -BIG;
  for(int i = tid; i < n; i += 256){ float v = p[i]; mn = fminf(mn,v); mx = fmaxf(mx,v); }
  __shared__ float smn[256], smx[256];
  smn[tid] = mn; smx[tid] = mx; __syncthreads();
  for(int s = 128; s > 0; s >>= 1){
    if(tid < s){ smn[tid] = fminf(smn[tid], smn[tid+s]); smx[tid] = fmaxf(smx[tid], smx[tid+s]); }
    __syncthreads();
  }
  if(tid == 0){ stats[blockIdx.x*2+0] = smn[0]; stats[blockIdx.x*2+1] = smx[0]; }
}

// ---------- pyr0 = (x - mn) / (mx - mn) ----------
__global__ void normalize_kernel(const float* __restrict__ x, const float* __restrict__ stats,
                                 float* __restrict__ out, int n){
  int i = blockIdx.x*256 + threadIdx.x;
  if(i >= n*NIMG) return;
  int img = i / n;
  float mn = stats[img*2+0], mx = stats[img*2+1];
  out[i] = (x[i] - mn) / (mx - mn);
}

// ---------- horizontal 6-tap, stride 2, reflect: in(H x W) -> tmp(H x W/2) ----------
__global__ void hconv_kernel(const float* __restrict__ in, float* __restrict__ tmp, int H, int W){
  const int Wout = W >> 1;
  const int per = H * Wout;
  int i = blockIdx.x*256 + threadIdx.x;
  if(i >= per*NIMG) return;
  int img = i / per;
  int r = i - img*per;
  int y = r / Wout, j = r - y*Wout;
  const float* p = in + (size_t)img*H*W + (size_t)y*W;
  const float wv[6] = {1.f/32.f, 5.f/32.f, 10.f/32.f, 10.f/32.f, 5.f/32.f, 1.f/32.f};
  float acc = 0.0f;
  #pragma unroll
  for(int b = 0; b < 6; b++){
    int q = 2*j - 2 + b;
    q = (q < 0) ? -q : ((q >= W) ? (2*W - 2 - q) : q);
    acc += wv[b] * p[q];
  }
  tmp[i] = acc;
}

// ---------- vertical 6-tap stride 2 via V_WMMA_F32_16X16X4_F32 ----------
// Out(16x16) = A(16x36) * X(36x16);  A[m][r] = w[r - 2m]  (constant band matrix),
// X rows are tmp rows S..S+35 (S = 2*i0 - 2) with reflect.  9 K=4 chunks.
// One wave per output tile; blockDim.x == 32 so EXEC is all ones for WMMA.
__global__ void vconv_wmma_kernel(const float* __restrict__ tmp, float* __restrict__ out,
                                  int Hin, int Wout, int Hout, int tilesX){
  const int lane = threadIdx.x;
  const int half = lane >> 4;      // 0: lanes 0-15, 1: lanes 16-31
  const int m    = lane & 15;
  const int img  = blockIdx.y;
  const int ty   = blockIdx.x / tilesX;
  const int tx   = blockIdx.x - ty*tilesX;
  const int i0 = ty*16, j0 = tx*16;
  const float* tp = tmp + (size_t)img*Hin*Wout;
  int col = j0 + m; if(col >= Wout) col = Wout - 1;   // clamped gather column
  const int S = 2*i0 - 2;
  v8f c = {};
  #pragma unroll
  for(int k = 0; k < 9; k++){
    const int kk = 4*k + 2*half;            // A layout: VGPR0 holds K=2*half, VGPR1 K=2*half+1
    v2f a = { wband(kk - 2*m), wband(kk + 1 - 2*m) };
    int g0 = S + kk;                        // B layout: VGPR0 row K=2*half, VGPR1 row K=2*half+1
    int g1 = g0 + 1;
    g0 = (g0 < 0) ? -g0 : ((g0 >= Hin) ? 2*Hin - 2 - g0 : g0);
    g1 = (g1 < 0) ? -g1 : ((g1 >= Hin) ? 2*Hin - 2 - g1 : g1);
    g0 = min(max(g0, 0), Hin - 1);
    g1 = min(max(g1, 0), Hin - 1);
    v2f bfrag = { tp[(size_t)g0*Wout + col], tp[(size_t)g1*Wout + col] };
    c = __builtin_amdgcn_wmma_f32_16x16x4_f32(false, a, false, bfrag, (short)0, c, false, false);
  }
  float* op = out + (size_t)img*Hout*Wout;
  const int cw = j0 + m;                    // C/D layout: VGPR v -> row v (+8 for lanes 16-31)
  #pragma unroll
  for(int v = 0; v < 8; v++){
    int row = i0 + v + 8*half;
    if(row < Hout && cw < Wout) op[(size_t)row*Wout + cw] = c[v];
  }
}

// ---------- d = |fine - nearest2x(coarse)| ----------
__global__ void diff_kernel(const float* __restrict__ fine, const float* __restrict__ coarse,
                            float* __restrict__ d, int s){
  const int per = s*s;
  int i = blockIdx.x*256 + threadIdx.x;
  if(i >= per*NIMG) return;
  int img = i / per;
  int r = i - img*per;
  int y = r / s, x = r - y*s;
  int hs = s >> 1;
  float a = fine[i];
  float b = coarse[(size_t)img*hs*hs + (size_t)(y>>1)*hs + (x>>1)];
  d[i] = fabsf(a - b);
}

// ---------- per-image: mn, 1/(mx-mn), (1-lm)^2 where lm = mean of 16x16 tile maxima ----------
__global__ void stats_kernel(const float* __restrict__ dimg, float* __restrict__ lstats, int s){
  const int tid = threadIdx.x;
  const int img = blockIdx.x;
  const float* p = dimg + (size_t)img*s*s;
  const int n = s*s;
  float mn = BIG, mx = -BIG;
  for(int i = tid; i < n; i += 256){ float v = p[i]; mn = fminf(mn,v); mx = fmaxf(mx,v); }
  __shared__ float smn[256], smx[256], ssum[256];
  smn[tid] = mn; smx[tid] = mx; __syncthreads();
  for(int r = 128; r > 0; r >>= 1){
    if(tid < r){ smn[tid] = fminf(smn[tid], smn[tid+r]); smx[tid] = fmaxf(smx[tid], smx[tid+r]); }
    __syncthreads();
  }
  // len(range(0, s-16, 16)) tiles per dim (last tile row/col skipped; 0 for s<=16)
  const int nt = (s > 16) ? (s - 1)/16 : 0;
  float sum = 0.0f;
  for(int t = tid; t < nt*nt; t += 256){
    int ty = t / nt, tx = t - ty*nt;
    const float* q = p + (size_t)(ty*16)*s + tx*16;
    float tm = -BIG;
    for(int yy = 0; yy < 16; yy++){
      const float* row = q + (size_t)yy*s;
      #pragma unroll
      for(int xx = 0; xx < 16; xx++) tm = fmaxf(tm, row[xx]);
    }
    sum += tm;
  }
  ssum[tid] = sum; __syncthreads();
  for(int r = 128; r > 0; r >>= 1){
    if(tid < r) ssum[tid] += ssum[tid+r];
    __syncthreads();
  }
  if(tid == 0){
    float gmn = smn[0], gmx = smx[0];
    float inv = 1.0f/(gmx - gmn);
    float sc = 1.0f;
    if(nt > 0){
      float lm = (ssum[0]/(float)(nt*nt) - gmn) * inv;   // mean tile max of normalized d
      float om = 1.0f - lm;
      sc = om*om;
    }
    lstats[img*3+0] = gmn; lstats[img*3+1] = inv; lstats[img*3+2] = sc;
  }
}

// ---------- fused: sum over i,c of bilinear-512 upsample of affine(d_i), then /3 ----------
__global__ void final_kernel(const float* __restrict__ ws, const float* __restrict__ lstats,
                             float* __restrict__ out, FinalArgs fa){
  int idx = blockIdx.x*256 + threadIdx.x;
  if(idx >= 8*512*512) return;
  int b = idx >> 18;
  int rem = idx & 262143;
  int y = rem >> 9;
  int x = rem & 511;
  float acc = 0.0f;
  for(int c = 0; c < 3; c++){
    int img = b*3 + c;
    #pragma unroll
    for(int i = 0; i < 7; i++){
      int s = 256 >> i;
      const float* dp = ws + fa.doff[i] + (long)img*s*s;
      const float* st = lstats + (i*NIMG + img)*3;
      float mn = st[0], inv = st[1], sc = st[2];
      float f = (float)s * (1.0f/512.0f);
      float fy = (y + 0.5f)*f - 0.5f;
      float fx = (x + 0.5f)*f - 0.5f;
      int y0f = (int)floorf(fy), x0f = (int)floorf(fx);
      float wy = fy - (float)y0f, wx = fx - (float)x0f;
      int y0 = min(max(y0f,   0), s-1), y1 = min(max(y0f+1, 0), s-1);
      int x0 = min(max(x0f,   0), s-1), x1 = min(max(x0f+1, 0), s-1);
      const float* r0 = dp + (long)y0*s;
      const float* r1 = dp + (long)y1*s;
      float v = (r0[x0]*(1.0f-wx) + r0[x1]*wx)*(1.0f-wy)
              + (r1[x0]*(1.0f-wx) + r1[x1]*wx)*wy;
      acc += (v - mn)*inv*sc;     // affine transform commutes with bilinear resize
    }
  }
  out[idx] = acc * (1.0f/3.0f);
}

extern "C" void kernel_launch(void* const* d_in, const int* in_sizes, int n_in,
                              void* d_out, int out_size, void* d_ws, size_t ws_size,
                              hipStream_t stream) {
  (void)in_sizes; (void)n_in; (void)out_size; (void)ws_size;
  const float* x = (const float*)d_in[0];     // [8,3,512,512] f32
  float* out = (float*)d_out;                 // [8,512,512] f32
  float* w   = (float*)d_ws;

  // workspace layout (floats): [0..48) x stats, [64..568) level stats, pyramid, diffs, tmp
  const long XSTATS = 0;
  const long LSTATS = 64;
  long off = 1024;
  long pyrOff[9];
  for(int l = 0; l < 9; l++){ pyrOff[l] = off; long s = 512 >> l; off += (long)NIMG*s*s; }
  long dOff[7];
  for(int i = 0; i < 7; i++){ dOff[i] = off; long s = 256 >> i; off += (long)NIMG*s*s; }
  long tmpOff = off;                           // reused H x W/2 buffer (max 512x256 per image)

  // 1) per-image min/max of x, then pyr0 = norm_pc(x)
  minmax_kernel<<<NIMG, 256, 0, stream>>>(x, w + XSTATS, 512*512);
  {
    int total = NIMG*512*512;
    normalize_kernel<<<(total+255)/256, 256, 0, stream>>>(x, w + XSTATS, w + pyrOff[0], 512*512);
  }

  // 2) Gaussian pyramid: separable 6-tap stride-2 (h pass VALU, v pass WMMA f32)
  for(int l = 1; l < 9; l++){
    int Hin = 512 >> (l-1), Win = Hin, Wout = Win >> 1, Hout = Hin >> 1;
    int total = NIMG*Hin*Wout;
    hconv_kernel<<<(total+255)/256, 256, 0, stream>>>(w + pyrOff[l-1], w + tmpOff, Hin, Win);
    int tilesX = (Wout + 15)/16, tilesY = (Hout + 15)/16;
    dim3 g(tilesX*tilesY, NIMG);
    vconv_wmma_kernel<<<g, 32, 0, stream>>>(w + tmpOff, w + pyrOff[l], Hin, Wout, Hout, tilesX);
  }

  // 3) center-surround diffs + per-image normalization stats
  for(int i = 0; i < 7; i++){
    int s = 256 >> i;
    int total = NIMG*s*s;
    diff_kernel<<<(total+255)/256, 256, 0, stream>>>(w + pyrOff[i+1], w + pyrOff[i+2], w + dOff[i], s);
    stats_kernel<<<NIMG, 256, 0, stream>>>(w + dOff[i], w + LSTATS + (long)i*NIMG*3, s);
  }

  // 4) fused bilinear-to-512 accumulate + channel mean
  FinalArgs fa; for(int i = 0; i < 7; i++) fa.doff[i] = dOff[i];
  int total = 8*512*512;
  final_kernel<<<(total+255)/256, 256, 0, stream>>>(w, w + LSTATS, out, fa);
}